// VWGCN_24232205484694
// MI455X (gfx1250) — compile-verified
//
#include <hip/hip_runtime.h>

typedef __attribute__((ext_vector_type(2))) float v2f;
typedef __attribute__((ext_vector_type(8))) float v8f;

#define N_NODES 2048
#define BATCH   32
#define DIM_C   64
#define DIM_O   64
#define EMB     16
#define KSUP    4

// ---------------------------------------------------------------------------
// CDNA5 async memory->LDS helpers (ASYNCcnt-tracked; ISA 15.18.3 op 98).
// Low 32 bits of a generic pointer to __shared__ give the LDS byte offset.
// ---------------------------------------------------------------------------
__device__ __forceinline__ void async_copy16(void* lds_dst, const void* gsrc) {
    unsigned int lds_off = (unsigned int)(unsigned long long)lds_dst;
    asm volatile("global_load_async_to_lds_b128 %0, %1, off"
                 :: "v"(lds_off), "v"(gsrc) : "memory");
}
__device__ __forceinline__ void async_wait_le6() {
    asm volatile("s_wait_asynccnt 6" ::: "memory");
}
__device__ __forceinline__ void async_wait_le0() {
    asm volatile("s_wait_asynccnt 0" ::: "memory");
}

// ---------------------------------------------------------------------------
// xT[n, b*64+c] = x[b, n, c]   (2048 x 2048 fp32, coalesced both sides)
// ---------------------------------------------------------------------------
__global__ __launch_bounds__(256) void k_transpose(const float* __restrict__ x,
                                                   float* __restrict__ xT) {
    int idx = blockIdx.x * 256 + threadIdx.x;     // over N * (B*C) = 2048*2048
    int n = idx >> 11;
    int r = idx & 2047;                           // b*64 + c
    int b = r >> 6, c = r & 63;
    xT[idx] = x[(b * N_NODES + n) * DIM_C + c];
}

// ---------------------------------------------------------------------------
// apt = softmax(relu(E E^T), axis=1); one block per row
// ---------------------------------------------------------------------------
__global__ __launch_bounds__(256) void k_apt(const float* __restrict__ E,
                                             float* __restrict__ apt) {
    __shared__ float red[256];
    const int n = blockIdx.x, t = threadIdx.x;
    float e[EMB];
#pragma unroll
    for (int d = 0; d < EMB; ++d) e[d] = E[n * EMB + d];

    float mx = -1e30f;
    for (int j = t; j < N_NODES; j += 256) {
        float s = 0.f;
#pragma unroll
        for (int d = 0; d < EMB; ++d) s += e[d] * E[j * EMB + d];
        s = fmaxf(s, 0.f);
        apt[n * N_NODES + j] = s;
        mx = fmaxf(mx, s);
    }
    red[t] = mx; __syncthreads();
    for (int off = 128; off > 0; off >>= 1) {
        if (t < off) red[t] = fmaxf(red[t], red[t + off]);
        __syncthreads();
    }
    mx = red[0]; __syncthreads();

    float sum = 0.f;
    for (int j = t; j < N_NODES; j += 256) {
        float v = __expf(apt[n * N_NODES + j] - mx);
        apt[n * N_NODES + j] = v;
        sum += v;
    }
    red[t] = sum; __syncthreads();
    for (int off = 128; off > 0; off >>= 1) {
        if (t < off) red[t] += red[t + off];
        __syncthreads();
    }
    const float inv = 1.f / red[0];
    for (int j = t; j < N_NODES; j += 256) apt[n * N_NODES + j] *= inv;
}

// ---------------------------------------------------------------------------
// C = alpha * (A @ B) + diag * I   for 2048x2048 fp32 matrices.
// Block tile 128(M) x 64(N); 8 waves; each wave owns a 16x64 strip
// (4 v8f accumulators) so one A fragment feeds 4 v_wmma_f32_16x16x4_f32.
// K staged in 32-wide chunks, double-buffered in LDS via async copies:
// issue chunk c+1, s_wait_asynccnt 6 (in-order => chunk c landed), barrier,
// compute chunk c while c+1 is in flight.
// A tile padded to stride 36 floats: 144 B rows (16B-aligned for B128 LDS
// writes) and 36/4=9 odd => conflict-free column reads.
// ---------------------------------------------------------------------------
#define AS_STRIDE 36

__global__ __launch_bounds__(256) void k_gemm2048(const float* __restrict__ A,
                                                  const float* __restrict__ B,
                                                  float* __restrict__ C,
                                                  float alpha, float diag) {
    __shared__ __align__(16) float As[2][128 * AS_STRIDE];
    __shared__ __align__(16) float Bs[2][32 * 64];
    const int t    = threadIdx.x;
    const int lane = t & 31;
    const int w    = t >> 5;
    const int wm   = w * 16;              // wave's 16-row strip within 128
    const int mblk = blockIdx.y * 128;
    const int nblk = blockIdx.x * 64;
    const int l16  = lane & 15;
    const int kq   = (lane >> 4) * 2;     // K 0/1 in lanes 0-15, K 2/3 in 16-31

    // per-thread staging coordinates (6 x b128 per thread per chunk)
    const int am  = t >> 1;               // A: row 0..127
    const int ak  = (t & 1) * 16;         // A: k sub-offset 0/16
    const int bk  = t >> 3;               // B: k row 0..31
    const int bn  = (t & 7) * 8;          // B: n sub-offset

    // prologue: async-stage chunk 0 into buffer 0
    {
        const float* ga = A + (size_t)(mblk + am) * N_NODES + ak;
        float* la = &As[0][am * AS_STRIDE + ak];
#pragma unroll
        for (int s = 0; s < 4; ++s) async_copy16(la + s * 4, ga + s * 4);
        const float* gb = B + (size_t)bk * N_NODES + nblk + bn;
        float* lb = &Bs[0][bk * 64 + bn];
        async_copy16(lb, gb);
        async_copy16(lb + 4, gb + 4);
    }

    v8f acc[4] = {{}, {}, {}, {}};
    for (int kb = 0; kb < N_NODES; kb += 32) {
        const int cur = (kb >> 5) & 1;
        if (kb + 32 < N_NODES) {   // stage next chunk into the other buffer
            const float* ga = A + (size_t)(mblk + am) * N_NODES + (kb + 32) + ak;
            float* la = &As[cur ^ 1][am * AS_STRIDE + ak];
#pragma unroll
            for (int s = 0; s < 4; ++s) async_copy16(la + s * 4, ga + s * 4);
            const float* gb = B + (size_t)(kb + 32 + bk) * N_NODES + nblk + bn;
            float* lb = &Bs[cur ^ 1][bk * 64 + bn];
            async_copy16(lb, gb);
            async_copy16(lb + 4, gb + 4);
            async_wait_le6();      // chunk kb fully landed (in-order completion)
        } else {
            async_wait_le0();
        }
        __syncthreads();

        const float* Asc = &As[cur][0];
        const float* Bsc = &Bs[cur][0];
#pragma unroll
        for (int kk = 0; kk < 32; kk += 4) {
            v2f a;
            a.x = Asc[(wm + l16) * AS_STRIDE + kk + kq];
            a.y = Asc[(wm + l16) * AS_STRIDE + kk + kq + 1];
#pragma unroll
            for (int c = 0; c < 4; ++c) {
                v2f b;
                b.x = Bsc[(kk + kq) * 64 + c * 16 + l16];
                b.y = Bsc[(kk + kq + 1) * 64 + c * 16 + l16];
                acc[c] = __builtin_amdgcn_wmma_f32_16x16x4_f32(
                    false, a, false, b, (short)0, acc[c], false, false);
            }
        }
        __syncthreads();           // reads done before next chunk overwrites
    }

#pragma unroll
    for (int c = 0; c < 4; ++c) {
#pragma unroll
        for (int j = 0; j < 8; ++j) {
            int gm = mblk + wm + j + (lane >> 4) * 8;
            int gn = nblk + c * 16 + l16;
            C[(size_t)gm * N_NODES + gn] =
                alpha * acc[c][j] + ((gm == gn) ? diag : 0.f);
        }
    }
}

// ---------------------------------------------------------------------------
// out[b,n,o] = E[n] . bias_pool[:,o]   (initializes out before accumulation)
// ---------------------------------------------------------------------------
__global__ __launch_bounds__(256) void k_bias(const float* __restrict__ E,
                                              const float* __restrict__ bias_pool,
                                              float* __restrict__ out) {
    int idx = blockIdx.x * 256 + threadIdx.x;     // over B*N*O
    int o = idx & 63;
    int n = (idx >> 6) & (N_NODES - 1);
    float s = 0.f;
#pragma unroll
    for (int d = 0; d < EMB; ++d) s += E[n * EMB + d] * bias_pool[d * DIM_O + o];
    out[idx] = s;
}

// ---------------------------------------------------------------------------
// Per-node contraction for one support k:
//   W_{n,k}[i,o] = sum_d E[n,d] * Wp[d,k,i,o]          (built in LDS, 64x64)
//   out[b,n,o]  += Y_k[n, b*64+i] @ W_{n,k}[i,o]       (32x64x64 WMMA GEMM)
// One block per node; Y row staged via async copies (stride 68: 272B rows,
// 16B-aligned, 68/4=17 odd => conflict-free column reads).
// ---------------------------------------------------------------------------
__global__ __launch_bounds__(256) void k_node_gemm(const float* __restrict__ Y,
                                                   const float* __restrict__ E,
                                                   const float* __restrict__ Wp,
                                                   float* __restrict__ out,
                                                   int ksup) {
    __shared__ __align__(16) float Ws[64 * 64];   // W_{n,k}, row-major [i][o]
    __shared__ __align__(16) float As[32 * 68];   // Y row of node n as (b, i)
    const int n = blockIdx.x;
    const int t = threadIdx.x;

    {   // async-stage A: row n of Y (2048 floats)
        int b = t >> 3, i8 = (t & 7) * 8;
        const float* yp = Y + (size_t)n * 2048 + b * 64 + i8;
        float* dst = &As[b * 68 + i8];
        async_copy16(dst, yp);
        async_copy16(dst + 4, yp + 4);
    }
    {   // build W_{n,k} in LDS (coalesced over o) while A is in flight
        float e[EMB];
#pragma unroll
        for (int d = 0; d < EMB; ++d) e[d] = E[n * EMB + d];
        for (int ee = t; ee < 64 * 64; ee += 256) {
            int i = ee >> 6, o = ee & 63;
            float s = 0.f;
#pragma unroll
            for (int d = 0; d < EMB; ++d)
                s += e[d] * Wp[((d * KSUP + ksup) * DIM_C + i) * DIM_O + o];
            Ws[ee] = s;
        }
    }
    async_wait_le0();
    __syncthreads();

    const int lane = t & 31, w = t >> 5;
    const int wm = (w & 1) * 16, wn = (w >> 1) * 16;
    const int l16 = lane & 15, kq = (lane >> 4) * 2;

    v8f acc = {};
#pragma unroll
    for (int k0 = 0; k0 < 64; k0 += 4) {
        v2f a, b;
        a.x = As[(wm + l16) * 68 + k0 + kq];
        a.y = As[(wm + l16) * 68 + k0 + kq + 1];
        b.x = Ws[(k0 + kq) * 64 + wn + l16];
        b.y = Ws[(k0 + kq + 1) * 64 + wn + l16];
        acc = __builtin_amdgcn_wmma_f32_16x16x4_f32(
            false, a, false, b, (short)0, acc, false, false);
    }
#pragma unroll
    for (int j = 0; j < 8; ++j) {
        int b = wm + j + (lane >> 4) * 8;
        int o = wn + l16;
        out[((size_t)b * N_NODES + n) * DIM_O + o] += acc[j];
    }
}

// ---------------------------------------------------------------------------
extern "C" void kernel_launch(void* const* d_in, const int* in_sizes, int n_in,
                              void* d_out, int out_size, void* d_ws, size_t ws_size,
                              hipStream_t stream) {
    const float* x   = (const float*)d_in[0];   // (32, 2048, 64)
    const float* E   = (const float*)d_in[1];   // (2048, 16)
    const float* lap = (const float*)d_in[2];   // (2048, 2048)
    const float* Wp  = (const float*)d_in[3];   // (16, 4, 64, 64)
    const float* bp  = (const float*)d_in[4];   // (16, 64)
    float* out = (float*)d_out;                 // (32, 2048, 64)

    // workspace: xT | T2 | apt | Ybuf  (4 x 16 MB fp32 = 64 MB)
    const size_t M = (size_t)N_NODES * N_NODES;
    float* xT  = (float*)d_ws;
    float* T2  = xT + M;
    float* apt = T2 + M;
    float* Yb  = apt + M;

    dim3 gemmGrid(N_NODES / 64, N_NODES / 128);

    k_transpose<<<(N_NODES * N_NODES) / 256, 256, 0, stream>>>(x, xT);
    k_apt<<<N_NODES, 256, 0, stream>>>(E, apt);
    // T2 = 2*lap@lap - I
    k_gemm2048<<<gemmGrid, 256, 0, stream>>>(lap, lap, T2, 2.0f, -1.0f);
    // out = bias
    k_bias<<<(BATCH * N_NODES * DIM_O) / 256, 256, 0, stream>>>(E, bp, out);
    // k = 0: identity support -> Y0 == xT
    k_node_gemm<<<N_NODES, 256, 0, stream>>>(xT, E, Wp, out, 0);
    // k = 1: lap
    k_gemm2048<<<gemmGrid, 256, 0, stream>>>(lap, xT, Yb, 1.0f, 0.0f);
    k_node_gemm<<<N_NODES, 256, 0, stream>>>(Yb, E, Wp, out, 1);
    // k = 2: T2
    k_gemm2048<<<gemmGrid, 256, 0, stream>>>(T2, xT, Yb, 1.0f, 0.0f);
    k_node_gemm<<<N_NODES, 256, 0, stream>>>(Yb, E, Wp, out, 2);
    // k = 3: adaptive support
    k_gemm2048<<<gemmGrid, 256, 0, stream>>>(apt, xT, Yb, 1.0f, 0.0f);
    k_node_gemm<<<N_NODES, 256, 0, stream>>>(Yb, E, Wp, out, 3);
}